// SmileMoENorm_87436944212181
// MI455X (gfx1250) — compile-verified
//
#include <hip/hip_runtime.h>
#include <math.h>
#include <stdint.h>

#define HIDDEN   1024
#define NEXP     8
#define EPS      1e-5f
#define BLOCK    256
#define WAVES    (BLOCK / 32)
#define GRID     2048

// native ext-vector float4 (HIP's float4 is a class type; builtins need a real vector)
typedef __attribute__((ext_vector_type(4))) float f4;
// builtin's expected element type: int __attribute__((vector_size(16)))
typedef int v4i __attribute__((vector_size(16)));

typedef __attribute__((address_space(1))) v4i glb_v4i;
typedef __attribute__((address_space(3))) v4i lds_v4i;
typedef __attribute__((address_space(3))) void lds_void;

__global__ __launch_bounds__(BLOCK) void moe_norm_kernel(
    const float* __restrict__ x,       // [N, 1024]
    const float* __restrict__ logits,  // [N, 8]
    const float* __restrict__ gamma,   // [8, 1024]
    const float* __restrict__ beta,    // [8, 1024]
    float* __restrict__ out,           // [N, 1024]
    int n_tokens)
{
    // 64 KB LDS: gamma then beta, f4-aligned
    __shared__ f4 lds_g[NEXP * HIDDEN / 4];   // 2048 f4
    __shared__ f4 lds_b[NEXP * HIDDEN / 4];   // 2048 f4

    const int tid  = threadIdx.x;
    const int lane = tid & 31;
    const int wave = tid >> 5;

    // ---- Stage gamma/beta into LDS via gfx1250 async global->LDS DMA ----
    {
        const f4* g4 = (const f4*)gamma;
        const f4* b4 = (const f4*)beta;
#if __has_builtin(__builtin_amdgcn_global_load_async_to_lds_b128)
        #pragma unroll
        for (int k = 0; k < (NEXP * HIDDEN / 4) / BLOCK; ++k) {   // 8 iters
            int i = tid + k * BLOCK;
            __builtin_amdgcn_global_load_async_to_lds_b128(
                (glb_v4i*)(g4 + i), (lds_v4i*)(lds_g + i), 0, 0);
            __builtin_amdgcn_global_load_async_to_lds_b128(
                (glb_v4i*)(b4 + i), (lds_v4i*)(lds_b + i), 0, 0);
        }
#else
        #pragma unroll
        for (int k = 0; k < (NEXP * HIDDEN / 4) / BLOCK; ++k) {
            int i = tid + k * BLOCK;
            unsigned goff = (unsigned)(i * 16);
            unsigned lg = (unsigned)(uintptr_t)(lds_void*)(lds_g + i);
            unsigned lb = (unsigned)(uintptr_t)(lds_void*)(lds_b + i);
            asm volatile("global_load_async_to_lds_b128 %0, %1, %2 offset:0"
                         :: "v"(lg), "v"(goff), "s"(gamma) : "memory");
            asm volatile("global_load_async_to_lds_b128 %0, %1, %2 offset:0"
                         :: "v"(lb), "v"(goff), "s"(beta) : "memory");
        }
#endif
#if __has_builtin(__builtin_amdgcn_s_wait_asynccnt)
        __builtin_amdgcn_s_wait_asynccnt(0);
#else
        asm volatile("s_wait_asynccnt 0" ::: "memory");
#endif
        __syncthreads();
    }

    const float inv_h = 1.0f / (float)HIDDEN;

    // one wave per token; 32 lanes x 8 f4 = 1024 floats held in registers
    for (int t = blockIdx.x * WAVES + wave; t < n_tokens; t += gridDim.x * WAVES) {
        const f4* xp = (const f4*)(x + (size_t)t * HIDDEN);
        f4 v[8];
        #pragma unroll
        for (int i = 0; i < 8; ++i)
            v[i] = __builtin_nontemporal_load(xp + lane + 32 * i);  // 512B/wave/ld, NT stream

        // mean & E[x^2] (partial per lane)
        float s = 0.f, ss = 0.f;
        #pragma unroll
        for (int i = 0; i < 8; ++i) {
            s += v[i].x + v[i].y + v[i].z + v[i].w;
            ss = fmaf(v[i].x, v[i].x, ss);
            ss = fmaf(v[i].y, v[i].y, ss);
            ss = fmaf(v[i].z, v[i].z, ss);
            ss = fmaf(v[i].w, v[i].w, ss);
        }
        // wave32 butterfly reduction
        #pragma unroll
        for (int m = 16; m > 0; m >>= 1) {
            s  += __shfl_xor(s,  m, 32);
            ss += __shfl_xor(ss, m, 32);
        }
        float mean = s * inv_h;
        float var  = fmaf(ss, inv_h, -mean * mean);
        float rstd = rsqrtf(var + EPS);

        // top-2 routing; softmax denominator cancels after renormalization:
        // w1 = exp(l1)/(exp(l0)+exp(l1)) = 1/(1+exp(l0-l1)), w0 = 1-w1
        const float* rl = logits + (size_t)t * NEXP;
        float l0 = -INFINITY, l1 = -INFINITY;
        int   e0 = 0,         e1 = 0;
        #pragma unroll
        for (int e = 0; e < NEXP; ++e) {
            float l = rl[e];
            if (l > l0)      { l1 = l0; e1 = e0; l0 = l; e0 = e; }
            else if (l > l1) { l1 = l;  e1 = e; }
        }
        float w1 = 1.0f / (1.0f + __expf(l0 - l1));
        float w0 = 1.0f - w1;

        const f4* g0 = lds_g + e0 * (HIDDEN / 4);
        const f4* g1 = lds_g + e1 * (HIDDEN / 4);
        const f4* b0 = lds_b + e0 * (HIDDEN / 4);
        const f4* b1 = lds_b + e1 * (HIDDEN / 4);
        f4* op = (f4*)(out + (size_t)t * HIDDEN);

        #pragma unroll
        for (int i = 0; i < 8; ++i) {
            int idx = lane + 32 * i;
            f4 G0 = g0[idx], G1 = g1[idx];   // ds_load_b128
            f4 B0 = b0[idx], B1 = b1[idx];
            f4 r;
            r.x = fmaf((v[i].x - mean) * rstd, fmaf(w0, G0.x, w1 * G1.x), fmaf(w0, B0.x, w1 * B1.x));
            r.y = fmaf((v[i].y - mean) * rstd, fmaf(w0, G0.y, w1 * G1.y), fmaf(w0, B0.y, w1 * B1.y));
            r.z = fmaf((v[i].z - mean) * rstd, fmaf(w0, G0.z, w1 * G1.z), fmaf(w0, B0.z, w1 * B1.z));
            r.w = fmaf((v[i].w - mean) * rstd, fmaf(w0, G0.w, w1 * G1.w), fmaf(w0, B0.w, w1 * B1.w));
            __builtin_nontemporal_store(r, op + idx);   // NT streaming store
        }
    }
}

extern "C" void kernel_launch(void* const* d_in, const int* in_sizes, int n_in,
                              void* d_out, int out_size, void* d_ws, size_t ws_size,
                              hipStream_t stream) {
    const float* x      = (const float*)d_in[0];
    const float* logits = (const float*)d_in[1];
    const float* gamma  = (const float*)d_in[2];
    const float* beta   = (const float*)d_in[3];
    float* out = (float*)d_out;

    int n_tokens = in_sizes[0] / HIDDEN;
    int grid = GRID;
    int need = (n_tokens + WAVES - 1) / WAVES;
    if (grid > need) grid = need;
    if (grid < 1) grid = 1;

    moe_norm_kernel<<<grid, BLOCK, 0, stream>>>(x, logits, gamma, beta, out, n_tokens);
}